// UserAggregator_20864951124275
// MI455X (gfx1250) — compile-verified
//
#include <hip/hip_runtime.h>
#include <math.h>

// ---------------------------------------------------------------------------
// Fused hyperbolic aggregator for MI455X (gfx1250, wave32).
//  - 1 workgroup = 16 batch rows (one WMMA M-tile), 256 threads = 8 waves.
//  - Thread mapping for vector ops: r = tid>>4 (row), j = tid&15, each thread
//    owns 4 consecutive floats of its row (D=64). Row reductions = 16-lane
//    xor-shuffle butterflies (masks 1,2,4,8 stay inside each wave32 half).
//  - Two fp32 GEMMs ([16x64]@[64x64]^T) via V_WMMA_F32_16X16X4_F32.
//  - Neighbor tensor streamed twice; 2nd pass hits L2 (per-WG tile = 256KB).
// ---------------------------------------------------------------------------

#define EPSF  1e-15f
#define MAXNF (1.0f - 1e-5f)

typedef __attribute__((ext_vector_type(2))) float v2f;
typedef __attribute__((ext_vector_type(8))) float v8f;

#if __has_builtin(__builtin_amdgcn_wmma_f32_16x16x4_f32)
#define HAVE_WMMA_F32X4 1
#endif

__device__ __forceinline__ float artanh_(float x) {
  x = fminf(fmaxf(x, -1.0f + 1e-7f), 1.0f - 1e-7f);
  return 0.5f * (log1pf(x) - log1pf(-x));
}
__device__ __forceinline__ float rsum16(float v) {
  v += __shfl_xor(v, 1, 32);
  v += __shfl_xor(v, 2, 32);
  v += __shfl_xor(v, 4, 32);
  v += __shfl_xor(v, 8, 32);
  return v;   // all 16 lanes of the half-wave hold the row sum
}
__device__ __forceinline__ float rmax16(float v) {
  v = fmaxf(v, __shfl_xor(v, 1, 32));
  v = fmaxf(v, __shfl_xor(v, 2, 32));
  v = fmaxf(v, __shfl_xor(v, 4, 32));
  v = fmaxf(v, __shfl_xor(v, 8, 32));
  return v;
}
__device__ __forceinline__ float dot4(const float4& a, const float4& b) {
  return a.x*b.x + a.y*b.y + a.z*b.z + a.w*b.w;
}
__device__ __forceinline__ float4 sc4(float s, const float4& a) {
  return make_float4(s*a.x, s*a.y, s*a.z, s*a.w);
}
__device__ __forceinline__ float4 ax_by(float a, const float4& x, float b, const float4& y) {
  return make_float4(a*x.x + b*y.x, a*x.y + b*y.y, a*x.z + b*y.z, a*x.w + b*y.w);
}

// C[16x64] = A[16x64] @ W[64x64]^T  (C[m][n] = sum_k A[m][k]*W[n][k])
// Waves 0..3 each produce one 16x16 N-tile with 16 K-steps of K=4.
// ISA fragment layouts (CDNA5 ISA §7.12.2):
//   A (16x4 f32, 2 VGPRs): lane<16 -> k=reg, lane>=16 -> k=reg+2, m=lane&15
//   B (4x16 f32, 2 VGPRs): symmetric, n=lane&15
//   C/D (16x16 f32, 8 VGPRs): m = reg + 8*(lane>=16), n = lane&15
__device__ __forceinline__ void wmma_gemm16x64(const float (*A)[68], const float (*W)[68],
                                               float (*C)[68], int tid) {
  const int wave = tid >> 5;
  if (wave >= 4) return;                 // whole waves branch -> EXEC all ones
  const int lane = tid & 31;
  const int l15  = lane & 15;
  const int half = lane >> 4;            // 0 or 1
  const int kh   = half << 1;            // 0 or 2
  const int n    = (wave << 4) + l15;    // output column of this lane
#if defined(HAVE_WMMA_F32X4)
  v8f acc = {};
  #pragma unroll
  for (int k0 = 0; k0 < 64; k0 += 4) {
    v2f a, b;
    a[0] = A[l15][k0 + kh];  a[1] = A[l15][k0 + kh + 1];
    b[0] = W[n  ][k0 + kh];  b[1] = W[n  ][k0 + kh + 1];
    // (neg_a, A, neg_b, B, c_mod, C, reuse_a, reuse_b)
    acc = __builtin_amdgcn_wmma_f32_16x16x4_f32(false, a, false, b, (short)0, acc,
                                                false, false);
  }
  #pragma unroll
  for (int v = 0; v < 8; ++v) C[(half << 3) + v][n] = acc[v];
#else
  float acc[8];
  #pragma unroll
  for (int v = 0; v < 8; ++v) acc[v] = 0.0f;
  for (int k = 0; k < 64; ++k) {
    const float bw = W[n][k];
    #pragma unroll
    for (int v = 0; v < 8; ++v) acc[v] += A[(half << 3) + v][k] * bw;
  }
  #pragma unroll
  for (int v = 0; v < 8; ++v) C[(half << 3) + v][n] = acc[v];
#endif
}

__global__ __launch_bounds__(256) void hyp_agg_kernel(
    const float* __restrict__ self, const float* __restrict__ ngh,
    const float* __restrict__ Wui,  const float* __restrict__ linW,
    const float* __restrict__ linb, float* __restrict__ out, int B)
{
  // Padded rows: 68 floats = 272 B = 17*16 -> float4-aligned, conflict-free
  __shared__ __align__(16) float sWbuf[64][68];   // W_ui, then lin_W
  __shared__ __align__(16) float sLU[16][68];     // logmap0(self)
  __shared__ __align__(16) float sUW[16][68];     // GEMM outputs (uw, then mx)
  __shared__ __align__(16) float sV [16][68];     // out1 (A-matrix of GEMM2)
  __shared__ float sY2[16][64];                   // |q|^2
  __shared__ float sPQ[16][64];                   // <p,q>
  __shared__ float sUQ[16][64];                   // <uw,q>
  __shared__ float sWB[16][64];                   // att*coef*B  per (r,s)
  __shared__ __align__(16) float sEB[64];         // expmap0(lin_b)
  __shared__ float sEBy2;
  __shared__ float rowX2[16], rowIL[16], rowCA[16], rowXN[16];

  const int tid = threadIdx.x;
  const int r   = tid >> 4;     // row in tile
  const int j   = tid & 15;     // lane-in-row
  const int b0  = blockIdx.x << 4;
  const int gr  = (b0 + r < B) ? (b0 + r) : (B - 1);

  // ---- stage W_ui ----
  for (int idx = tid; idx < 4096; idx += 256)
    sWbuf[idx >> 6][idx & 63] = Wui[idx];

  // ---- eb = proj(expmap0(lin_b)), done by wave 0 ----
  if (tid < 32) {
    const float bl = linb[tid], bh = linb[tid + 32];
    float s2 = bl*bl + bh*bh;
    s2 += __shfl_xor(s2, 1, 32);  s2 += __shfl_xor(s2, 2, 32);
    s2 += __shfl_xor(s2, 4, 32);  s2 += __shfl_xor(s2, 8, 32);
    s2 += __shfl_xor(s2, 16, 32);
    const float nb = fmaxf(sqrtf(s2), EPSF);
    const float th = tanhf(nb);
    float sc = th / nb;
    float pn = fmaxf(th, EPSF);
    const float ps = (pn > MAXNF) ? (MAXNF / pn) : 1.0f;
    sc *= ps; pn *= ps;
    sEB[tid]      = bl * sc;
    sEB[tid + 32] = bh * sc;
    if (tid == 0) sEBy2 = pn * pn;
  }

  // ---- phase P: load p, logmap0(self) -> sLU ----
  const float4 p4 = *reinterpret_cast<const float4*>(self + (size_t)gr*64 + 4*j);
  const float x2 = rsum16(dot4(p4, p4));
  {
    const float np = fmaxf(sqrtf(x2), EPSF);
    const float ls = artanh_(np) / np;
    *reinterpret_cast<float4*>(&sLU[r][4*j]) = sc4(ls, p4);
    if (j == 0) { rowX2[r] = x2; rowIL[r] = fmaxf(1.0f - x2, EPSF); }
  }
  __syncthreads();

  // ---- GEMM1: uw = lu @ W_ui^T ----
  wmma_gemm16x64(sLU, sWbuf, sUW, tid);
  __syncthreads();

  const float4 uw4 = *reinterpret_cast<const float4*>(&sUW[r][4*j]);
  const float* qb = ngh + ((size_t)gr * 64) * 64;   // row base, S=D=64

  // ---- pass 1: stream neighbors, 3 dots per (r,s) ----
  for (int s = 0; s < 64; ++s) {
    const float4 q4 = *reinterpret_cast<const float4*>(qb + s*64 + 4*j);
    const float y2 = rsum16(dot4(q4, q4));
    const float pq = rsum16(dot4(p4, q4));
    const float uq = rsum16(dot4(uw4, q4));
    if (j == 0) { sY2[r][s] = y2; sPQ[r][s] = pq; sUQ[r][s] = uq; }
  }
  // stage lin_W (sWbuf free after GEMM1 sync)
  for (int idx = tid; idx < 4096; idx += 256)
    sWbuf[idx >> 6][idx & 63] = linW[idx];
  __syncthreads();

  // ---- softmax over s + logmap(p,q) scalar coefficients ----
  {
    const float invl = rowIL[r];             // = 2/lam = max(1-x2, eps)
    float lg[4], yy[4], pp[4], e[4];
    float mloc = -3.0e38f;
    #pragma unroll
    for (int i = 0; i < 4; ++i) {
      const int s = 4*j + i;
      yy[i] = sY2[r][s]; pp[i] = sPQ[r][s];
      const float uq = sUQ[r][s];
      const float nq = fmaxf(sqrtf(yy[i]), EPSF);
      lg[i] = artanh_(nq) / nq * uq * 0.015625f;   // mean over D=64
      mloc = fmaxf(mloc, lg[i]);
    }
    const float m = rmax16(mloc);
    float es = 0.0f;
    #pragma unroll
    for (int i = 0; i < 4; ++i) { e[i] = __expf(lg[i] - m); es += e[i]; }
    const float invZ = 1.0f / rsum16(es);
    float caP = 0.0f;
    #pragma unroll
    for (int i = 0; i < 4; ++i) {
      const int s = 4*j + i;
      const float att = e[i] * invZ;
      // mobius_add(-p,q) = A*(-p) + Bc*q ; xy = -<p,q>
      const float den = fmaxf(1.0f - 2.0f*pp[i] + x2*yy[i], EPSF);
      const float A   = (1.0f - 2.0f*pp[i] + yy[i]) / den;
      const float Bc  = (1.0f - x2) / den;
      const float ns2 = A*A*x2 - 2.0f*A*Bc*pp[i] + Bc*Bc*yy[i];
      const float ns  = fmaxf(sqrtf(ns2), EPSF);
      const float w   = att * invl * artanh_(ns) / ns;   // att * (2/lam)*artanh/|.|
      sWB[r][s] = w * Bc;
      caP += w * A;
    }
    const float ca = rsum16(caP);
    if (j == 0) rowCA[r] = ca;
  }
  __syncthreads();

  // ---- pass 2 (L2-resident re-read): agg, expmap(agg,p), mobius_add(p,.) ----
  {
    float4 acc = make_float4(0.f, 0.f, 0.f, 0.f);
    for (int s = 0; s < 64; ++s) {
      const float4 q4 = *reinterpret_cast<const float4*>(qb + s*64 + 4*j);
      const float wb = sWB[r][s];
      acc.x += wb*q4.x; acc.y += wb*q4.y; acc.z += wb*q4.z; acc.w += wb*q4.w;
    }
    const float ca = rowCA[r];
    const float4 agg = ax_by(1.0f, acc, -ca, p4);
    const float nu2 = rsum16(dot4(agg, agg));
    const float nu  = fmaxf(sqrtf(nu2), EPSF);
    const float invl = rowIL[r];
    const float tt  = tanhf(nu / invl) / nu;     // tanh(lam*nu/2)/nu
    const float4 sec = sc4(tt, agg);
    const float y2  = tt*tt*nu2;
    const float xy  = rsum16(dot4(p4, sec));
    const float den = fmaxf(1.0f + 2.0f*xy + x2*y2, EPSF);
    float4 h = ax_by((1.0f + 2.0f*xy + y2)/den, p4, (1.0f - x2)/den, sec);
    float nh = fmaxf(sqrtf(rsum16(dot4(h, h))), EPSF);
    const float ps = (nh > MAXNF) ? (MAXNF/nh) : 1.0f;
    h = sc4(ps, h); nh *= ps;
    const float y2h = nh*nh;
    const float xy2 = rsum16(dot4(p4, h));
    const float den2 = fmaxf(1.0f + 2.0f*xy2 + x2*y2h, EPSF);
    const float4 o1 = ax_by((1.0f + 2.0f*xy2 + y2h)/den2, p4, (1.0f - x2)/den2, h);
    const float xn = fmaxf(sqrtf(rsum16(dot4(o1, o1))), EPSF);
    *reinterpret_cast<float4*>(&sV[r][4*j]) = o1;
    if (j == 0) rowXN[r] = xn;
  }
  __syncthreads();

  // ---- GEMM2: mx = out1 @ lin_W^T ----
  wmma_gemm16x64(sV, sWbuf, sUW, tid);
  __syncthreads();

  // ---- mobius_matvec tail, +bias, relu-in-tangent, store ----
  {
    const float4 mx4 = *reinterpret_cast<const float4*>(&sUW[r][4*j]);
    const float mxn = fmaxf(sqrtf(rsum16(dot4(mx4, mx4))), EPSF);
    const float xn  = rowXN[r];
    const float sc  = tanhf(mxn / xn * artanh_(xn)) / mxn;
    float4 o2 = sc4(sc, mx4);
    float no2 = fmaxf(fabsf(sc) * mxn, EPSF);
    float ps = (no2 > MAXNF) ? (MAXNF/no2) : 1.0f;
    o2 = sc4(ps, o2); no2 *= ps;
    const float x2o = no2*no2;
    const float y2e = sEBy2;
    const float4 eb4 = *reinterpret_cast<const float4*>(&sEB[4*j]);
    const float xye = rsum16(dot4(o2, eb4));
    const float den = fmaxf(1.0f + 2.0f*xye + x2o*y2e, EPSF);
    float4 o3 = ax_by((1.0f + 2.0f*xye + y2e)/den, o2, (1.0f - x2o)/den, eb4);
    float n3 = fmaxf(sqrtf(rsum16(dot4(o3, o3))), EPSF);
    ps = (n3 > MAXNF) ? (MAXNF/n3) : 1.0f;
    o3 = sc4(ps, o3); n3 *= ps;
    const float lsc = artanh_(n3) / n3;
    float4 v = make_float4(fmaxf(lsc*o3.x, 0.f), fmaxf(lsc*o3.y, 0.f),
                           fmaxf(lsc*o3.z, 0.f), fmaxf(lsc*o3.w, 0.f));
    const float nv = fmaxf(sqrtf(rsum16(dot4(v, v))), EPSF);
    const float th = tanhf(nv);
    float osc = th / nv;
    ps = (th > MAXNF) ? (MAXNF/th) : 1.0f;
    osc *= ps;
    if (b0 + r < B)
      *reinterpret_cast<float4*>(out + (size_t)(b0 + r)*64 + 4*j) = sc4(osc, v);
  }
}

extern "C" void kernel_launch(void* const* d_in, const int* in_sizes, int n_in,
                              void* d_out, int out_size, void* d_ws, size_t ws_size,
                              hipStream_t stream) {
  const float* self = (const float*)d_in[0];
  const float* ngh  = (const float*)d_in[1];
  const float* Wui  = (const float*)d_in[2];
  const float* linW = (const float*)d_in[3];
  const float* linb = (const float*)d_in[4];
  float* out = (float*)d_out;
  const int D = 64;                       // fixed by reference
  const int B = in_sizes[0] / D;          // 16384
  dim3 grid((B + 15) / 16), block(256);
  hipLaunchKernelGGL(hyp_agg_kernel, grid, block, 0, stream,
                     self, ngh, Wui, linW, linb, out, B);
}